// BaseESN_68994354643274
// MI455X (gfx1250) — compile-verified
//
#include <hip/hip_runtime.h>
#include <math.h>

// ESN forward: T=4096, INPUT=64, RES=4096, OUT=32, all fp32.
//   U = inputs @ W_in^T            (WMMA f32 16x16x4 GEMM)
//   x_t = tanh(u_t + W @ x_{t-1})  (persistent kernel, grid barrier per step)
//   Y = X @ W_out                  (WMMA f32 16x16x4 GEMM)
// Workspace: buf[4096*4096] f32 (U overwritten in place by X) + 1 u32 counter.

#define T_STEPS 4096
#define RES     4096
#define INPUT   64
#define OUT     32

#define SCAN_NBLK 256
#define SCAN_TPB  256

typedef float v2f __attribute__((ext_vector_type(2)));
typedef float v8f __attribute__((ext_vector_type(8)));

// ---------------------------------------------------------------------------
// Stage A: U[t][r] = sum_k inputs[t][k] * W_in[r][k]
// One wave per 16x16 tile of U. 256x256 tiles = 65536 waves.
// ---------------------------------------------------------------------------
__global__ void __launch_bounds__(256)
esn_u_gemm(const float* __restrict__ inp, const float* __restrict__ Win,
           float* __restrict__ U) {
  const int lane = threadIdx.x & 31;
  const int wave = (blockIdx.x * blockDim.x + threadIdx.x) >> 5;
  const int tm = wave >> 8;    // t-tile   0..255
  const int tn = wave & 255;   // res-tile 0..255
  const int half = lane >> 4;  // 0: K=0,1   1: K=2,3
  const int lid  = lane & 15;

  const float* aptr = inp + (tm * 16 + lid) * INPUT + half * 2;  // A: M=lid
  const float* bptr = Win + (tn * 16 + lid) * INPUT + half * 2;  // B: N=lid

  v8f acc = {};
#pragma unroll
  for (int k0 = 0; k0 < INPUT; k0 += 4) {
    v2f a = *(const v2f*)(aptr + k0);
    v2f b = *(const v2f*)(bptr + k0);
    acc = __builtin_amdgcn_wmma_f32_16x16x4_f32(
        false, a, false, b, (short)0, acc, false, false);
  }
  // C layout: VGPR v -> M = v + 8*half, N = lid
  float* cbase = U + (size_t)(tm * 16 + half * 8) * RES + tn * 16 + lid;
#pragma unroll
  for (int v = 0; v < 8; ++v) cbase[(size_t)v * RES] = acc[v];
}

// ---------------------------------------------------------------------------
// Init: X[0] = 0 (row 0 of buf), barrier counter = 0. Runs after stage A.
// ---------------------------------------------------------------------------
__global__ void esn_init(float* __restrict__ buf, unsigned int* __restrict__ cnt) {
  int i = blockIdx.x * blockDim.x + threadIdx.x;
  if (i < RES) buf[i] = 0.0f;
  if (i == 0) *cnt = 0u;
}

// ---------------------------------------------------------------------------
// Stage B: persistent scan. buf[t] = tanh(buf[t] + W @ buf[t-1]), t = 1..4095.
// 256 blocks x 256 threads; block owns 16 rows; 16 lanes per row.
// Grid barrier: monotonic agent-scope atomic counter in workspace.
// ---------------------------------------------------------------------------
__device__ __forceinline__ void grid_sync(unsigned int* cnt, unsigned int target) {
  __threadfence();          // release this block's buf[t] stores device-wide
  __syncthreads();
  if (threadIdx.x == 0) {
    __hip_atomic_fetch_add(cnt, 1u, __ATOMIC_ACQ_REL, __HIP_MEMORY_SCOPE_AGENT);
    while (__hip_atomic_load(cnt, __ATOMIC_ACQUIRE, __HIP_MEMORY_SCOPE_AGENT) < target) {
      __builtin_amdgcn_s_sleep(1);
    }
  }
  __syncthreads();
  __threadfence();          // acquire other blocks' stores
}

__global__ void __launch_bounds__(SCAN_TPB)
esn_scan(const float* __restrict__ W, float* __restrict__ buf,
         unsigned int* __restrict__ cnt) {
  const int rowLocal = threadIdx.x >> 4;                 // 0..15
  const int seg      = threadIdx.x & 15;                 // 0..15
  const int row      = blockIdx.x * 16 + rowLocal;       // 0..4095
  const float* __restrict__ wrow = W + (size_t)row * RES;

  unsigned int target = 0;
  for (int t = 1; t < T_STEPS; ++t) {
    const float* __restrict__ xprev = buf + (size_t)(t - 1) * RES;
    float acc = 0.0f;
    // 16 lanes per row, float4 each -> 64 contiguous floats per iteration.
#pragma unroll 8
    for (int i = 0; i < RES / 64; ++i) {
      const int j = i * 64 + seg * 4;
      const float4 w4 = *(const float4*)(wrow + j);
      const float4 x4 = *(const float4*)(xprev + j);
      acc += w4.x * x4.x + w4.y * x4.y + w4.z * x4.z + w4.w * x4.w;
    }
    // reduce across the 16 lanes of this row
    acc += __shfl_xor(acc, 8, 16);
    acc += __shfl_xor(acc, 4, 16);
    acc += __shfl_xor(acc, 2, 16);
    acc += __shfl_xor(acc, 1, 16);
    if (seg == 0) {
      float* xcur = buf + (size_t)t * RES + row;
      *xcur = tanhf(*xcur + acc);   // *xcur holds u[t][row]
    }
    target += SCAN_NBLK;
    grid_sync(cnt, target);
  }
}

// ---------------------------------------------------------------------------
// Stage C: Y[t][o] = sum_r X[t][r] * W_out[r][o]. 256x2 tiles = 512 waves.
// ---------------------------------------------------------------------------
__global__ void __launch_bounds__(1024)
esn_out_gemm(const float* __restrict__ X, const float* __restrict__ Wout,
             float* __restrict__ Y) {
  const int lane = threadIdx.x & 31;
  const int wave = (blockIdx.x * blockDim.x + threadIdx.x) >> 5;
  const int tm = wave >> 1;   // 0..255
  const int tn = wave & 1;    // 0..1
  const int half = lane >> 4;
  const int lid  = lane & 15;

  const float* aptr = X + (size_t)(tm * 16 + lid) * RES + half * 2;
  const float* bptr = Wout + (size_t)(half * 2) * OUT + (tn * 16 + lid);

  v8f acc = {};
#pragma unroll 8
  for (int k0 = 0; k0 < RES; k0 += 4) {
    v2f a = *(const v2f*)(aptr + k0);
    v2f b;
    b.x = bptr[(size_t)(k0 + 0) * OUT];
    b.y = bptr[(size_t)(k0 + 1) * OUT];
    acc = __builtin_amdgcn_wmma_f32_16x16x4_f32(
        false, a, false, b, (short)0, acc, false, false);
  }
  float* cbase = Y + (size_t)(tm * 16 + half * 8) * OUT + tn * 16 + lid;
#pragma unroll
  for (int v = 0; v < 8; ++v) cbase[(size_t)v * OUT] = acc[v];
}

// ---------------------------------------------------------------------------
extern "C" void kernel_launch(void* const* d_in, const int* in_sizes, int n_in,
                              void* d_out, int out_size, void* d_ws, size_t ws_size,
                              hipStream_t stream) {
  const float* inputs = (const float*)d_in[0];   // [T, INPUT]
  const float* W_in   = (const float*)d_in[1];   // [RES, INPUT]
  const float* W      = (const float*)d_in[2];   // [RES, RES]
  const float* W_out  = (const float*)d_in[3];   // [RES, OUT]
  float* Y = (float*)d_out;                      // [T, OUT]

  float* buf = (float*)d_ws;                                   // [T, RES] f32
  unsigned int* cnt = (unsigned int*)((char*)d_ws +
                      (size_t)T_STEPS * RES * sizeof(float));  // barrier ctr

  // Stage A: U = inputs @ W_in^T   (65536 waves)
  esn_u_gemm<<<8192, 256, 0, stream>>>(inputs, W_in, buf);
  // Zero X[0] and the barrier counter (must run after stage A wrote row 0)
  esn_init<<<16, 256, 0, stream>>>(buf, cnt);
  // Stage B: sequential scan, persistent kernel with per-step grid barrier
  esn_scan<<<SCAN_NBLK, SCAN_TPB, 0, stream>>>(W, buf, cnt);
  // Stage C: Y = X @ W_out   (512 waves)
  esn_out_gemm<<<16, 1024, 0, stream>>>(buf, W_out, Y);
}